// GatedI2TLoss_60078002536928
// MI455X (gfx1250) — compile-verified
//
#include <hip/hip_runtime.h>
#include <hip/hip_bf16.h>

#define NSAMP 131072
#define NCLS  1000
#define DFEAT 512
#define NREP  8   // L2-resident privatized replicas of the scatter target

typedef __attribute__((ext_vector_type(2))) float v2f;
typedef __attribute__((ext_vector_type(8))) float v8f;

// ---------------------------------------------------------------- zero ws
__global__ void zero_ws_kernel(float* __restrict__ ws, long n) {
  long i = (long)blockIdx.x * blockDim.x + threadIdx.x;
  long stride = (long)gridDim.x * blockDim.x;
  for (; i < n; i += stride) ws[i] = 0.0f;
}

// -------------------------------------------- fused argmax + gated scatter
// One wave32 per sample row. 8 waves per 256-thread block.
__global__ __launch_bounds__(256) void argmax_scatter_kernel(
    const float* __restrict__ logits, const float* __restrict__ img,
    const unsigned char* __restrict__ gate,
    float* __restrict__ sums, float* __restrict__ counts) {
  const int lane = threadIdx.x & 31;
  const int wave = threadIdx.x >> 5;
  const int row  = blockIdx.x * 8 + wave;

  const unsigned char g = gate[row];
  const float* frow = img + (size_t)row * DFEAT;
  if (g) {
    // 2KB feature row: one 64B line per lane, warmed while argmax streams.
    __builtin_prefetch(frow + lane * 16, 0, 3);
  }

  const float* lrow = logits + (size_t)row * NCLS;
  float best = -3.402823466e38f;
  int   bidx = 0;
#pragma unroll
  for (int it = 0; it < 8; ++it) {
    int base = it * 128 + lane * 4;
    if (base + 3 < NCLS) {                 // tail lanes (idx >= 1000) drop out cleanly
      float4 v = *(const float4*)(lrow + base);
      if (v.x > best) { best = v.x; bidx = base;     }
      if (v.y > best) { best = v.y; bidx = base + 1; }
      if (v.z > best) { best = v.z; bidx = base + 2; }
      if (v.w > best) { best = v.w; bidx = base + 3; }
    }
  }
  // wave32 argmax reduction; lowest index wins ties (jnp.argmax semantics)
  for (int off = 16; off > 0; off >>= 1) {
    float ov = __shfl_xor(best, off, 32);
    int   oi = __shfl_xor(bidx, off, 32);
    if (ov > best || (ov == best && oi < bidx)) { best = ov; bidx = oi; }
  }

  if (!g) return;
  const int rep = blockIdx.x & (NREP - 1);
  if (lane == 0) atomicAdd(counts + rep * NCLS + bidx, 1.0f);
  float* srow = sums + ((size_t)rep * NCLS + bidx) * DFEAT;
#pragma unroll
  for (int it = 0; it < DFEAT / 128; ++it) {
    int j = it * 128 + lane * 4;
    float4 v = *(const float4*)(frow + j);
    atomicAdd(srow + j + 0, v.x);
    atomicAdd(srow + j + 1, v.y);
    atomicAdd(srow + j + 2, v.z);
    atomicAdd(srow + j + 3, v.w);
  }
}

// ------------------------------- WMMA phase: dots[c] = <sums[c,:],text[c,:]>
// One full wave per 16-class tile. D-dim accumulated via V_WMMA_F32_16X16X4_F32;
// the needed row-wise dots are the diagonal of the 16x16 accumulator.
__global__ __launch_bounds__(32) void dots_wmma_kernel(
    const float* __restrict__ sums, const float* __restrict__ counts,
    const float* __restrict__ text, float* __restrict__ accum) {
  const int lane = threadIdx.x;
  const int c0 = blockIdx.x * 16;
  const int m  = lane & 15;            // A-row / B-col handled by this lane
  const int c  = c0 + m;
  const bool validc = c < NCLS;

  float cnt = 0.f;
  if (validc)
    for (int r = 0; r < NREP; ++r) cnt += counts[r * NCLS + c];

  // 16x4 f32 A layout: lanes 0-15 hold K=0,1; lanes 16-31 hold K=2,3 (M=lane%16).
  // 4x16 f32 B mirrors it: lanes hold (K=koff, K=koff+1) for column N=lane%16.
  const int koff = (lane < 16) ? 0 : 2;

  v8f acc = {};
  for (int d = 0; d < DFEAT; d += 4) {
    v2f a = {0.f, 0.f};
    v2f b = {0.f, 0.f};
    if (validc) {
      size_t si = (size_t)c * DFEAT + d + koff;
      float s0 = 0.f, s1 = 0.f;
#pragma unroll
      for (int r = 0; r < NREP; ++r) {
        const float* p = sums + (size_t)r * NCLS * DFEAT;
        s0 += p[si];
        s1 += p[si + 1];
      }
      a[0] = s0;       a[1] = s1;
      b[0] = text[si]; b[1] = text[si + 1];
    }
    // uniform control flow here: EXEC is all-1s as WMMA requires
    acc = __builtin_amdgcn_wmma_f32_16x16x4_f32(
        /*neg_a=*/false, a, /*neg_b=*/false, b,
        /*c_mod=*/(short)0, acc, /*reuse_a=*/false, /*reuse_b=*/false);
  }

  // Diagonal of 16x16 f32 C: lanes 0-7 -> M=lane @ VGPR=lane;
  //                          lanes 24-31 -> M=lane-16 @ VGPR=lane-24.
  const bool hasdiag = (lane < 8) || (lane >= 24);
  const int  vidx = (lane < 8) ? lane : ((lane >= 24) ? (lane - 24) : 0);
  float diag;
  switch (vidx) {
    case 0: diag = acc[0]; break; case 1: diag = acc[1]; break;
    case 2: diag = acc[2]; break; case 3: diag = acc[3]; break;
    case 4: diag = acc[4]; break; case 5: diag = acc[5]; break;
    case 6: diag = acc[6]; break; default: diag = acc[7]; break;
  }

  float dot = 0.f, pres = 0.f;
  if (hasdiag && validc && cnt > 0.f) {
    pres = 1.f;
    dot  = diag / cnt;   // cnt >= 1 when present, so max(cnt,1) == cnt
  }
  for (int off = 16; off > 0; off >>= 1) {
    dot  += __shfl_xor(dot,  off, 32);
    pres += __shfl_xor(pres, off, 32);
  }
  if (lane == 0) {
    atomicAdd(accum + 0, dot);
    atomicAdd(accum + 1, pres);
  }
}

// ---------------------------------------------------------------- finalize
__global__ void finalize_kernel(const float* __restrict__ accum,
                                float* __restrict__ out) {
  float total = accum[0];
  float uniq  = accum[1];
  out[0] = (uniq > 0.f) ? (total / uniq) : 0.f;
}

extern "C" void kernel_launch(void* const* d_in, const int* in_sizes, int n_in,
                              void* d_out, int out_size, void* d_ws, size_t ws_size,
                              hipStream_t stream) {
  (void)in_sizes; (void)n_in; (void)out_size; (void)ws_size;
  const float*         logits = (const float*)d_in[0];
  const float*         img    = (const float*)d_in[1];
  const float*         text   = (const float*)d_in[2];
  const unsigned char* gate   = (const unsigned char*)d_in[3]; // jax bool = 1 byte

  float* ws     = (float*)d_ws;
  float* sums   = ws;                                   // NREP*NCLS*DFEAT
  float* counts = sums + (size_t)NREP * NCLS * DFEAT;   // NREP*NCLS
  float* accum  = counts + NREP * NCLS;                 // 2 floats

  long zn = (long)NREP * NCLS * DFEAT + NREP * NCLS + 2;
  zero_ws_kernel<<<2048, 256, 0, stream>>>(ws, zn);
  argmax_scatter_kernel<<<NSAMP / 8, 256, 0, stream>>>(logits, img, gate, sums, counts);
  dots_wmma_kernel<<<(NCLS + 15) / 16, 32, 0, stream>>>(sums, counts, text, accum);
  finalize_kernel<<<1, 1, 0, stream>>>(accum, (float*)d_out);
}